// MoEGating_44762149159145
// MI455X (gfx1250) — compile-verified
//
#include <hip/hip_runtime.h>
#include <hip/hip_bf16.h>
#include <stdint.h>

// ---------------------------------------------------------------------------
// MoE top-1 gating + capacity-limited dispatch for MI455X (gfx1250, wave32).
// Memory-bound (~390 MB @ 23.3 TB/s ~= 17us). CDNA5 paths used:
//   - global_load_async_to_lds_b128 / s_wait_asynccnt (ASYNCcnt) row streaming
//   - wave32 ballot-based order-preserving per-expert ranking
// ---------------------------------------------------------------------------

#define N_EXPERTS 64
#define HIDDEN    3072
#define CHUNK     256   // tokens per ranking chunk / block

// --- Kernel 1: per-token softmax score + argmax (one wave per token) -------
__global__ void k_gating(const float* __restrict__ logits,
                         int* __restrict__ expertIdx,
                         float* __restrict__ rawScore,
                         int T) {
    int lane = threadIdx.x & 31;
    int wv   = threadIdx.x >> 5;
    int t    = blockIdx.x * 8 + wv;
    if (t >= T) return;

    const float* row = logits + (size_t)t * N_EXPERTS;
    float l0 = row[lane];
    float l1 = row[lane + 32];

    float m; int mi;
    if (l0 >= l1) { m = l0; mi = lane; } else { m = l1; mi = lane + 32; }
    #pragma unroll
    for (int off = 16; off > 0; off >>= 1) {
        float om = __shfl_xor(m, off, 32);
        int   oi = __shfl_xor(mi, off, 32);
        if (om > m || (om == m && oi < mi)) { m = om; mi = oi; }
    }
    float s = __expf(l0 - m) + __expf(l1 - m);
    #pragma unroll
    for (int off = 16; off > 0; off >>= 1) s += __shfl_xor(s, off, 32);

    if (lane == 0) {
        expertIdx[t] = mi;
        rawScore[t]  = 1.0f / s;   // probs[argmax] = exp(0)/sum
    }
}

// --- Kernel 2a: per-chunk order-preserving ranks + chunk histograms --------
__global__ void k_rank(const int* __restrict__ expertIdx,
                       int* __restrict__ rankInChunk,
                       int* __restrict__ chunkHist) {
    __shared__ int hist[8][N_EXPERTS];   // [wave][expert]
    int tid  = threadIdx.x;
    int t    = blockIdx.x * CHUNK + tid;
    int e    = expertIdx[t];
    int lane = tid & 31;
    int wv   = tid >> 5;

    // same-expert lane mask via 6 ballots over expert-id bits (wave32)
    unsigned same = 0xFFFFFFFFu;
    #pragma unroll
    for (int b = 0; b < 6; ++b) {
        unsigned bal = (unsigned)__ballot((e >> b) & 1);
        same &= ((e >> b) & 1) ? bal : ~bal;
    }
    unsigned ltmask = (lane == 0) ? 0u : ((1u << lane) - 1u);
    int rankLane = __popc(same & ltmask);
    bool leader  = (same & ltmask) == 0u;

    int* h = &hist[0][0];
    h[tid] = 0; h[tid + 256] = 0;
    __syncthreads();
    if (leader) hist[wv][e] = __popc(same);
    __syncthreads();
    if (tid < N_EXPERTS) {               // exclusive scan across 8 waves
        int run = 0;
        #pragma unroll
        for (int w = 0; w < 8; ++w) { int v = hist[w][tid]; hist[w][tid] = run; run += v; }
        chunkHist[blockIdx.x * N_EXPERTS + tid] = run;
    }
    __syncthreads();
    rankInChunk[t] = hist[wv][e] + rankLane;
}

// --- Kernel 2b: scan chunk histograms, emit expert_counts ------------------
__global__ void k_scan(const int* __restrict__ chunkHist,
                       int* __restrict__ chunkBase,
                       float* __restrict__ countOut,
                       const int* __restrict__ capPtr,
                       int nChunks) {
    int e   = threadIdx.x;               // 64 threads
    int cap = capPtr[0];
    int run = 0;
    for (int c = 0; c < nChunks; ++c) {
        int v = chunkHist[c * N_EXPERTS + e];
        chunkBase[c * N_EXPERTS + e] = run;
        run += v;
    }
    countOut[e] = (float)(run < cap ? run : cap);
}

// --- init slot->token map to -1 --------------------------------------------
__global__ void k_init(int* __restrict__ slotToToken, int size) {
    int i = blockIdx.x * 256 + threadIdx.x;
    if (i < size) slotToToken[i] = -1;
}

// --- Kernel 2c: finalize per-token outputs + inverse map -------------------
__global__ void k_finalize(const int* __restrict__ expertIdx,
                           const int* __restrict__ rankInChunk,
                           const int* __restrict__ chunkBase,
                           const float* __restrict__ rawScore,
                           const int* __restrict__ capPtr,
                           int* __restrict__ slotToToken,
                           float* __restrict__ scoreOut,
                           float* __restrict__ assignOut,
                           float* __restrict__ slotOut,
                           int T) {
    int t = blockIdx.x * 256 + threadIdx.x;
    if (t >= T) return;
    int e   = expertIdx[t];
    int loc = chunkBase[(t >> 8) * N_EXPERTS + e] + rankInChunk[t];
    int cap = capPtr[0];
    bool keep = loc < cap;
    int slot  = e * cap + loc;

    scoreOut[t]  = keep ? rawScore[t] : 0.0f;
    assignOut[t] = (float)e;
    slotOut[t]   = keep ? (float)slot : -1.0f;
    if (keep) slotToToken[slot] = t;
}

// --- Kernel 3: scale-and-scatter rows via async global->LDS ---------------
// One block per output row. 3072 fp32 = 768 float4 = 3 b128 per thread.
__global__ void k_scatter(const float* __restrict__ act,
                          const int* __restrict__ slotToToken,
                          const float* __restrict__ scoreOut,
                          float* __restrict__ moeOut) {
    __shared__ float buf[HIDDEN];
    int tid   = threadIdx.x;
    int slot  = blockIdx.x;
    int token = slotToToken[slot];
    float4* orow = (float4*)(moeOut + (size_t)slot * HIDDEN);

    if (token < 0) {
        float4 z = {0.f, 0.f, 0.f, 0.f};
        #pragma unroll
        for (int c = 0; c < 3; ++c) orow[c * 256 + tid] = z;
        return;
    }

    float s = scoreOut[token];
    const float* grow = act + (size_t)token * HIDDEN;

    // CDNA5 async copy: global -> LDS, tracked by ASYNCcnt.
    #pragma unroll
    for (int c = 0; c < 3; ++c) {
        int idx = c * 1024 + tid * 4;                       // float index
        unsigned ldsAddr = (unsigned)(unsigned long long)(&buf[idx]); // low 32b of flat ptr = LDS offset
        unsigned long long ga = (unsigned long long)(grow + idx);
        asm volatile("global_load_async_to_lds_b128 %0, %1, off"
                     :: "v"(ldsAddr), "v"(ga) : "memory");
    }
    asm volatile("s_wait_asynccnt 0" ::: "memory");

    #pragma unroll
    for (int c = 0; c < 3; ++c) {
        int idx = c * 1024 + tid * 4;
        float4 v = *(const float4*)&buf[idx];
        v.x *= s; v.y *= s; v.z *= s; v.w *= s;
        orow[c * 256 + tid] = v;
    }
}

// ---------------------------------------------------------------------------
extern "C" void kernel_launch(void* const* d_in, const int* in_sizes, int n_in,
                              void* d_out, int out_size, void* d_ws, size_t ws_size,
                              hipStream_t stream) {
    const float* act    = (const float*)d_in[0];
    const float* logits = (const float*)d_in[1];
    const int*   capPtr = (const int*)d_in[2];

    const int T = in_sizes[1] / N_EXPERTS;                 // 16384 tokens
    const int size = (out_size - 3 * T - N_EXPERTS) / HIDDEN; // padded dispatch rows
    const int nChunks = (T + CHUNK - 1) / CHUNK;

    // d_out layout: moe_input | score | expert_assignment | mapped_slots | expert_counts
    float* moeOut    = (float*)d_out;
    float* scoreOut  = moeOut + (size_t)size * HIDDEN;
    float* assignOut = scoreOut + T;
    float* slotOut   = assignOut + T;
    float* countOut  = slotOut + T;

    // workspace layout
    int*   expertIdx   = (int*)d_ws;
    int*   rankInChunk = expertIdx + T;
    float* rawScore    = (float*)(rankInChunk + T);
    int*   chunkHist   = (int*)(rawScore + T);
    int*   chunkBase   = chunkHist + nChunks * N_EXPERTS;
    int*   slotToToken = chunkBase + nChunks * N_EXPERTS;

    k_gating<<<(T + 7) / 8, 256, 0, stream>>>(logits, expertIdx, rawScore, T);
    k_rank<<<nChunks, CHUNK, 0, stream>>>(expertIdx, rankInChunk, chunkHist);
    k_scan<<<1, N_EXPERTS, 0, stream>>>(chunkHist, chunkBase, countOut, capPtr, nChunks);
    k_init<<<(size + 255) / 256, 256, 0, stream>>>(slotToToken, size);
    k_finalize<<<(T + 255) / 256, 256, 0, stream>>>(expertIdx, rankInChunk, chunkBase,
                                                    rawScore, capPtr, slotToToken,
                                                    scoreOut, assignOut, slotOut, T);
    k_scatter<<<size, 256, 0, stream>>>(act, slotToToken, scoreOut, moeOut);
}